// simpleNet_23862838297450
// MI455X (gfx1250) — compile-verified
//
#include <hip/hip_runtime.h>
#include <hip/hip_bf16.h>
#include <math.h>

typedef __attribute__((ext_vector_type(2))) float v2f;
typedef __attribute__((ext_vector_type(8))) float v8f;

#define CC 64  // channels

// ---------------------------------------------------------------------------
// Node-side GEMM: out[N,64] = in[N,64] @ W[64,64] (+ bias), using
// V_WMMA_F32_16X16X4_F32. One wave computes a 16-row x 64-col strip.
// A 16x4 frag: lane(0..15)=M, VGPR0/1 = K0/K1, lanes 16..31 carry K2/K3.
// B 4x16 frag: lane(0..15)=N, VGPR0/1 = K0/K1, lanes 16..31 carry K2/K3.
// C/D 16x16:   lane(0..15)=N, VGPR v = row v, lanes 16..31 -> row v+8.
// ---------------------------------------------------------------------------
__global__ void node_gemm64(const float* __restrict__ in,
                            const float* __restrict__ W,
                            const float* __restrict__ bias,
                            float* __restrict__ out, int n_rows) {
  const int wave   = threadIdx.x >> 5;
  const int lane   = threadIdx.x & 31;
  const int lhalf  = lane >> 4;      // 0 or 1
  const int l      = lane & 15;
  const int rowBase = (blockIdx.x * (blockDim.x >> 5) + wave) * 16;
  if (rowBase >= n_rows) return;     // wave-uniform exit; EXEC stays full

  v8f acc0 = {}, acc1 = {}, acc2 = {}, acc3 = {};
  const int arow = (rowBase + l < n_rows) ? (rowBase + l) : (n_rows - 1);

#pragma unroll
  for (int kb = 0; kb < 16; ++kb) {
    const int k0 = kb * 4 + lhalf * 2;     // this lane-half's K pair
    v2f a;
    a.x = in[(long long)arow * CC + k0];
    a.y = in[(long long)arow * CC + k0 + 1];

    v2f b0, b1, b2, b3;
    const float* wp = W + (long long)k0 * CC + l;
    b0.x = wp[0];        b0.y = wp[CC];
    b1.x = wp[16];       b1.y = wp[CC + 16];
    b2.x = wp[32];       b2.y = wp[CC + 32];
    b3.x = wp[48];       b3.y = wp[CC + 48];

    acc0 = __builtin_amdgcn_wmma_f32_16x16x4_f32(false, a, false, b0, (short)0, acc0, false, false);
    acc1 = __builtin_amdgcn_wmma_f32_16x16x4_f32(false, a, false, b1, (short)0, acc1, false, false);
    acc2 = __builtin_amdgcn_wmma_f32_16x16x4_f32(false, a, false, b2, (short)0, acc2, false, false);
    acc3 = __builtin_amdgcn_wmma_f32_16x16x4_f32(false, a, false, b3, (short)0, acc3, false, false);
  }

  v8f* accs[4] = {&acc0, &acc1, &acc2, &acc3};
#pragma unroll
  for (int ct = 0; ct < 4; ++ct) {
    const int col = ct * 16 + l;
    const float bv = bias ? bias[col] : 0.0f;
#pragma unroll
    for (int v = 0; v < 8; ++v) {
      const int row = rowBase + v + lhalf * 8;
      if (row < n_rows)
        out[(long long)row * CC + col] = (*accs[ct])[v] + bv;
    }
  }
}

// ---------------------------------------------------------------------------
// h[i] = x[i] (residual base), vectorized float4
// ---------------------------------------------------------------------------
__global__ void copy_f4(const float4* __restrict__ in, float4* __restrict__ out,
                        long long n4) {
  long long i = (long long)blockIdx.x * blockDim.x + threadIdx.x;
  if (i < n4) out[i] = in[i];
}

// ---------------------------------------------------------------------------
// Edge pass: msg = sigmoid(Af[d]+Bf[s]) * softplus(As[d]+Bs[s]); h[d] += msg.
// One thread handles 4 channels of one edge (float4 gathers from L2).
// ---------------------------------------------------------------------------
__global__ void edge_pass(const long long* __restrict__ src,
                          const long long* __restrict__ dst,
                          const float* __restrict__ Af, const float* __restrict__ Bf,
                          const float* __restrict__ As, const float* __restrict__ Bs,
                          float* __restrict__ h, long long nE) {
  long long tid = (long long)blockIdx.x * blockDim.x + threadIdx.x;
  long long total = nE * (CC / 4);
  if (tid >= total) return;
  long long e = tid >> 4;                 // 16 threads per edge
  int c4 = (int)(tid & 15);               // float4 slot
  long long d = dst[e];
  long long s = src[e];
  long long doff = d * CC + c4 * 4;
  long long soff = s * CC + c4 * 4;

  float4 fa = *(const float4*)(Af + doff);
  float4 fb = *(const float4*)(Bf + soff);
  float4 ga = *(const float4*)(As + doff);
  float4 gb = *(const float4*)(Bs + soff);

  float f[4] = {fa.x + fb.x, fa.y + fb.y, fa.z + fb.z, fa.w + fb.w};
  float g[4] = {ga.x + gb.x, ga.y + gb.y, ga.z + gb.z, ga.w + gb.w};

#pragma unroll
  for (int i = 0; i < 4; ++i) {
    float sig = 1.0f / (1.0f + __expf(-f[i]));
    float sp  = (g[i] > 20.0f) ? g[i] : log1pf(__expf(g[i]));
    atomicAdd(&h[doff + i], sig * sp);
  }
}

// ---------------------------------------------------------------------------
// Readout: accum[0] += surf[n] * dot(h[n], Wlin); accum[1] += surf[n]
// ---------------------------------------------------------------------------
__global__ void zero2(float* accum) {
  if (threadIdx.x < 2) accum[threadIdx.x] = 0.0f;
}

__global__ void readout(const float* __restrict__ h, const float* __restrict__ surf,
                        const float* __restrict__ Wlin, float* __restrict__ accum,
                        int n) {
  int nIdx = blockIdx.x * blockDim.x + threadIdx.x;
  float num = 0.0f, den = 0.0f;
  if (nIdx < n) {
    float dotv = 0.0f;
#pragma unroll
    for (int c = 0; c < CC; ++c) dotv += h[(long long)nIdx * CC + c] * Wlin[c];
    float sf = surf[nIdx];
    num = sf * dotv;
    den = sf;
  }
  // wave32 butterfly reduction
#pragma unroll
  for (int off = 16; off > 0; off >>= 1) {
    num += __shfl_xor(num, off, 32);
    den += __shfl_xor(den, off, 32);
  }
  if ((threadIdx.x & 31) == 0) {
    atomicAdd(&accum[0], num);
    atomicAdd(&accum[1], den);
  }
}

__global__ void finalize(const float* __restrict__ accum,
                         const float* __restrict__ blin,
                         float* __restrict__ out, int n) {
  if (threadIdx.x == 0 && blockIdx.x == 0)
    out[0] = (accum[0] + (float)n * blin[0]) / accum[1];
}

// ---------------------------------------------------------------------------
extern "C" void kernel_launch(void* const* d_in, const int* in_sizes, int n_in,
                              void* d_out, int out_size, void* d_ws, size_t ws_size,
                              hipStream_t stream) {
  const float*     x    = (const float*)d_in[0];
  const long long* ei   = (const long long*)d_in[1];
  const float*     surf = (const float*)d_in[2];
  const float*     Wf1  = (const float*)d_in[3];
  const float*     bf1  = (const float*)d_in[4];
  const float*     Ws1  = (const float*)d_in[5];
  const float*     bs1  = (const float*)d_in[6];
  const float*     Wf2  = (const float*)d_in[7];
  const float*     bf2  = (const float*)d_in[8];
  const float*     Ws2  = (const float*)d_in[9];
  const float*     bs2  = (const float*)d_in[10];
  const float*     Wlin = (const float*)d_in[11];
  const float*     blin = (const float*)d_in[12];
  float*           out  = (float*)d_out;

  const int       N = in_sizes[0] / CC;
  const long long E = in_sizes[1] / 2;
  const long long* src = ei;
  const long long* dst = ei + E;

  const size_t NC = (size_t)N * CC;
  float* Af = (float*)d_ws;
  float* As = Af + NC;
  float* Bf = As + NC;
  float* Bs = Bf + NC;
  float* h1 = Bs + NC;
  float* h2 = h1 + NC;
  float* accum = h2 + NC;

  // launch geometry
  const int gemmBlk = 256;                       // 8 waves
  const int rowTiles = (N + 15) / 16;
  const int gemmGrid = (rowTiles + 7) / 8;
  const long long n4 = NC / 4;
  const int copyGrid = (int)((n4 + 255) / 256);
  const long long edgeThreads = E * (CC / 4);
  const int edgeGrid = (int)((edgeThreads + 255) / 256);
  const int roGrid = (N + 255) / 256;

  // ---------------- layer 1 ----------------
  node_gemm64<<<gemmGrid, gemmBlk, 0, stream>>>(x, Wf1,            bf1,     Af, N); // top half of Wf1
  node_gemm64<<<gemmGrid, gemmBlk, 0, stream>>>(x, Wf1 + CC * CC,  nullptr, Bf, N); // bottom half
  node_gemm64<<<gemmGrid, gemmBlk, 0, stream>>>(x, Ws1,            bs1,     As, N);
  node_gemm64<<<gemmGrid, gemmBlk, 0, stream>>>(x, Ws1 + CC * CC,  nullptr, Bs, N);
  copy_f4<<<copyGrid, 256, 0, stream>>>((const float4*)x, (float4*)h1, n4);
  edge_pass<<<edgeGrid, 256, 0, stream>>>(src, dst, Af, Bf, As, Bs, h1, E);

  // ---------------- layer 2 ----------------
  node_gemm64<<<gemmGrid, gemmBlk, 0, stream>>>(h1, Wf2,           bf2,     Af, N);
  node_gemm64<<<gemmGrid, gemmBlk, 0, stream>>>(h1, Wf2 + CC * CC, nullptr, Bf, N);
  node_gemm64<<<gemmGrid, gemmBlk, 0, stream>>>(h1, Ws2,           bs2,     As, N);
  node_gemm64<<<gemmGrid, gemmBlk, 0, stream>>>(h1, Ws2 + CC * CC, nullptr, Bs, N);
  copy_f4<<<copyGrid, 256, 0, stream>>>((const float4*)h1, (float4*)h2, n4);
  edge_pass<<<edgeGrid, 256, 0, stream>>>(src, dst, Af, Bf, As, Bs, h2, E);

  // ---------------- readout ----------------
  zero2<<<1, 32, 0, stream>>>(accum);
  readout<<<roGrid, 256, 0, stream>>>(h2, surf, Wlin, accum, N);
  finalize<<<1, 32, 0, stream>>>(accum, blin, out, N);
}